// LearnableQuantization_24197845745917
// MI455X (gfx1250) — compile-verified
//
#include <hip/hip_runtime.h>
#include <hip/hip_bf16.h>

// LearnableQuantization training forward for MI455X (gfx1250).
//
// Shapes (from reference):
//   inp       (16,1,1024,8,8)        = 1,048,576 f32   -> flat index e, channel ch = e % 64
//   resize    (3,8,8)                = 192 f32
//   alpha     (8,8), beta (8,8), deviation (8,8)
//   grid_base (8,8,257)              = 16,448 f32
//   u         (16,1,1024,8,8,256)    = 268,435,456 f32  (1 GiB -- the streaming input)
//   i         scalar int
// Outputs (concatenated in d_out): out [nElem] f32, mean [64] f32, nzeros [1] f32.
//
// T == 1.0 exactly  =>  z = exp((log pi + u)/T) = pi * exp(u).
//
// Strategy: u is streamed once via the Tensor Data Mover (TENSOR_LOAD_TO_LDS),
// double-buffered per wave (TENSORcnt is per-wave and in-order, so no barriers).
// All sigmoid/softmax math stays in f32 on VALU/TRANS. LDS rows padded
// 1024B -> 1040B via the TDM pad feature to avoid bank conflicts.

#define KBINS 256
#define TILE_ELEMS 16                   // elements (rows of u) per wave tile
#define ROW_BYTES (KBINS * 4)           // 1024
#define ROW_STRIDE 1040                 // padded LDS row stride (=65*16, 16B aligned)
#define TILE_LDS_BYTES (TILE_ELEMS * ROW_STRIDE)  // 16640
#define WAVES_PER_WG 8
#define LDS_TOTAL (WAVES_PER_WG * 2 * TILE_LDS_BYTES)  // 266,240 B < 320 KiB/WGP

#if defined(__has_builtin)
#if __has_builtin(__builtin_amdgcn_tensor_load_to_lds)
#define HAVE_TDM 1
#endif
#endif
#ifndef HAVE_TDM
#define HAVE_TDM 0
#endif

typedef unsigned int v4u_t __attribute__((ext_vector_type(4)));
typedef int v8i_t __attribute__((ext_vector_type(8)));
typedef int v4i_t __attribute__((ext_vector_type(4)));

__device__ __forceinline__ void wait_tensorcnt_le1() {
#if HAVE_TDM
#if __has_builtin(__builtin_amdgcn_s_wait_tensorcnt)
  __builtin_amdgcn_s_wait_tensorcnt(1);
#else
  asm volatile("s_wait_tensorcnt 0x1" ::: "memory");
#endif
#endif
}
__device__ __forceinline__ void wait_tensorcnt_0() {
#if HAVE_TDM
#if __has_builtin(__builtin_amdgcn_s_wait_tensorcnt)
  __builtin_amdgcn_s_wait_tensorcnt(0);
#else
  asm volatile("s_wait_tensorcnt 0x0" ::: "memory");
#endif
#endif
}

// Issue (or synchronously emulate) a 16-row x 1KiB tile load of u into LDS,
// with 16B padding after every 1024B row.
__device__ __forceinline__ void load_u_tile(const float* __restrict__ u,
                                            unsigned long long tileByte,
                                            unsigned int ldsAddr,   // LDS byte offset (TDM)
                                            char* ldsPtr,           // generic ptr (fallback)
                                            int lane, int nElem) {
#if HAVE_TDM
  unsigned long long ga = (unsigned long long)(size_t)u + tileByte;
  v4u_t g0;
  g0[0] = 1u;                                   // count=1, user mode, no gather
  g0[1] = ldsAddr;                              // lds_addr (bytes)
  g0[2] = (unsigned int)(ga & 0xffffffffull);   // global_addr[31:0]
  g0[3] = (unsigned int)((ga >> 32) & 0x01ffffffull) | (2u << 30); // addr[56:32], type=2
  v8i_t g1;
  // data_size=2 (4B), pad_enable=1, pad_interval=7 (256 DWORDs), pad_amount=3 (4 DWORDs)
  g1[0] = (int)((2u << 16) | (1u << 20) | (7u << 22) | (3u << 25));
  g1[1] = (int)(((unsigned)KBINS & 0xffffu) << 16);            // tensor_dim0 lo16 = 256
  g1[2] = (int)(((unsigned)nElem & 0xffffu) << 16);            // tensor_dim1 lo16
  g1[3] = (int)((((unsigned)nElem >> 16) & 0xffffu) | ((unsigned)KBINS << 16)); // dim1 hi16 | tile_dim0=256
  g1[4] = TILE_ELEMS;                                          // tile_dim1 = 16 rows
  g1[5] = KBINS;                                               // tensor_dim0_stride = 256 elems
  g1[6] = 0;
  g1[7] = 0;
  v4i_t gz4 = {0, 0, 0, 0};
  v8i_t gz8 = {0, 0, 0, 0, 0, 0, 0, 0};
  // 6-arg form (clang-23 / therock-10.0 headers): (v4u, v8i, v4i, v4i, v8i, cpol)
  __builtin_amdgcn_tensor_load_to_lds(g0, g1, gz4, gz4, gz8, 0);
#else
  // Synchronous fallback: coalesced float4 copy with the same padded layout.
  const char* gsrc = (const char*)u + tileByte;
#pragma unroll
  for (int r = 0; r < TILE_ELEMS; ++r) {
    const float4* s = (const float4*)(gsrc + r * ROW_BYTES) + lane;
    float4* d = (float4*)(ldsPtr + r * ROW_STRIDE) + lane;
    d[0] = s[0];
    d[32] = s[32];
  }
#endif
}

__device__ __forceinline__ void compute_tile(int t, const char* __restrict__ ubuf, int lane,
                                             const float* __restrict__ inp,
                                             const float* __restrict__ rz,
                                             const float* __restrict__ alpha,
                                             const float* __restrict__ beta,
                                             const float* __restrict__ dev,
                                             const float* __restrict__ gridb,
                                             float* __restrict__ out) {
#pragma unroll 1
  for (int el = 0; el < TILE_ELEMS; ++el) {
    const int e = t * TILE_ELEMS + el;
    const int ch = e & 63;
    const float a = alpha[ch];
    const float bt = beta[ch];
    const float dv = dev[ch];
    const float rv = rz[ch];
    const float x = inp[e] / rv;
    const float invd = __builtin_amdgcn_rcpf(dv);
    // lane l covers bins k in [8l, 8l+8); needs grid boundaries 8l..8l+8
    const float* gr = gridb + ch * 257 + lane * 8;
    float gb[9], cdf[9];
#pragma unroll
    for (int j = 0; j < 9; ++j) gb[j] = gr[j];
#pragma unroll
    for (int j = 0; j < 9; ++j) {
      // cdf = sigmoid((grid - x)/dev),  grid = gb*alpha + beta
      const float tt = (__builtin_fmaf(gb[j], a, bt) - x) * invd;
      const float en = __expf(-tt);
      cdf[j] = __builtin_amdgcn_rcpf(1.0f + en);
    }
    const float cdf0 = __shfl(cdf[0], 0);
    const float cdfK = __shfl(cdf[8], 31);
    const float rden = __builtin_amdgcn_rcpf(cdfK - cdf0 + 2.56e-7f); // NOISE*K
    // 8 u-values for this lane: two 16B LDS loads (bank-conflict free via padding)
    const float4* up = (const float4*)(ubuf + el * ROW_STRIDE + lane * 32);
    const float4 u0 = up[0];
    const float4 u1 = up[1];
    const float uv[8] = {u0.x, u0.y, u0.z, u0.w, u1.x, u1.y, u1.z, u1.w};
    float sg = 0.0f, s1 = 0.0f;
#pragma unroll
    for (int j = 0; j < 8; ++j) {
      const float pi = (cdf[j + 1] - cdf[j] + 1e-9f) * rden;
      const float z = pi * __expf(uv[j]);   // T==1: exp((log pi + u)/T) == pi*exp(u)
      sg = __builtin_fmaf(z, gb[j], sg);    // sum z * grid_base
      s1 += z;                              // sum z
    }
#pragma unroll
    for (int o = 16; o > 0; o >>= 1) {
      sg += __shfl_xor(sg, o);
      s1 += __shfl_xor(s1, o);
    }
    if (lane == 0) {
      // sum(z*grid)/sum(z) * resize, grid = gb*a + bt
      out[e] = (a * sg + bt * s1) * __builtin_amdgcn_rcpf(s1) * rv;
    }
  }
}

__global__ void lq_main_kernel(const float* __restrict__ inp, const float* __restrict__ resize,
                               const float* __restrict__ alpha, const float* __restrict__ beta,
                               const float* __restrict__ dev, const float* __restrict__ gridb,
                               const float* __restrict__ u, const int* __restrict__ ip,
                               float* __restrict__ out, int nElem, int nTiles) {
  extern __shared__ char smem[];
  const int lane = (int)(threadIdx.x & 31u);
  const int wave = __builtin_amdgcn_readfirstlane((int)(threadIdx.x >> 5));
  const int i = ip[0];
  const float* rz = resize + i * 64;

  char* myLds = smem + wave * 2 * TILE_LDS_BYTES;
  const unsigned int ldsBase = (unsigned int)(size_t)smem + (unsigned int)(wave * 2 * TILE_LDS_BYTES);
  const unsigned int bufAddr[2] = {ldsBase, ldsBase + TILE_LDS_BYTES};
  char* bufPtr[2] = {myLds, myLds + TILE_LDS_BYTES};

  const int stride = (int)gridDim.x * WAVES_PER_WG;
  int t = (int)blockIdx.x * WAVES_PER_WG + wave;
  if (t < nTiles)
    load_u_tile(u, (unsigned long long)t * (TILE_ELEMS * ROW_BYTES), bufAddr[0], bufPtr[0], lane, nElem);
  int buf = 0;
  for (; t < nTiles; t += stride) {
    const int tn = t + stride;
    if (tn < nTiles) {
      load_u_tile(u, (unsigned long long)tn * (TILE_ELEMS * ROW_BYTES), bufAddr[buf ^ 1],
                  bufPtr[buf ^ 1], lane, nElem);
      wait_tensorcnt_le1();   // oldest (current tile) complete; next still in flight
    } else {
      wait_tensorcnt_0();
    }
    asm volatile("" ::: "memory");  // keep LDS reads after the wait
    compute_tile(t, bufPtr[buf], lane, inp, rz, alpha, beta, dev, gridb, out);
    buf ^= 1;
  }
}

// Deterministic per-channel mean(|inp/resize|): one block per channel, tree reduce.
__global__ void lq_mean_kernel(const float* __restrict__ inp, const float* __restrict__ resize,
                               const int* __restrict__ ip, float* __restrict__ out,
                               int nElem, int nPerCh) {
  __shared__ float red[256];
  const int ch = (int)blockIdx.x;
  const int i = ip[0];
  const float inv = 1.0f / resize[i * 64 + ch];
  float s = 0.0f;
  for (int j = (int)threadIdx.x; j < nPerCh; j += 256) s += fabsf(inp[j * 64 + ch] * inv);
  red[threadIdx.x] = s;
  __syncthreads();
  for (int w = 128; w > 0; w >>= 1) {
    if ((int)threadIdx.x < w) red[threadIdx.x] += red[threadIdx.x + w];
    __syncthreads();
  }
  if (threadIdx.x == 0) out[nElem + ch] = red[0] * (1.0f / (float)nPerCh);
  if (blockIdx.x == 0 && threadIdx.x == 1) out[nElem + 64] = 0.0f;  // nzeros
}

extern "C" void kernel_launch(void* const* d_in, const int* in_sizes, int n_in,
                              void* d_out, int out_size, void* d_ws, size_t ws_size,
                              hipStream_t stream) {
  const float* inp = (const float*)d_in[0];
  const float* resize = (const float*)d_in[1];
  const float* alpha = (const float*)d_in[2];
  const float* beta = (const float*)d_in[3];
  const float* dev = (const float*)d_in[4];
  const float* gridb = (const float*)d_in[5];
  const float* u = (const float*)d_in[6];
  const int* ip = (const int*)d_in[7];
  float* out = (float*)d_out;

  const int nElem = in_sizes[0];          // 1,048,576
  const int nPerCh = nElem / 64;          // 16,384
  const int nTiles = nElem / TILE_ELEMS;  // 65,536

  int blocks = 1024;
  if (blocks * WAVES_PER_WG > nTiles) blocks = (nTiles + WAVES_PER_WG - 1) / WAVES_PER_WG;

  lq_main_kernel<<<blocks, 256, LDS_TOTAL, stream>>>(inp, resize, alpha, beta, dev, gridb, u, ip,
                                                     out, nElem, nTiles);
  lq_mean_kernel<<<64, 256, 0, stream>>>(inp, resize, ip, out, nElem, nPerCh);
}